// QueryAndGroup_6176162972231
// MI455X (gfx1250) — compile-verified
//
#include <hip/hip_runtime.h>
#include <hip/hip_bf16.h>

// Problem constants (from reference setup_inputs)
#define BS 2
#define NP 16384        // parents
#define MQ 4096         // queries
#define CF 128          // feature channels
#define KK 32           // neighbors
#define CH (3 + CF)     // output channels (USE_XYZ)
#define R2 0.01f        // radius^2

typedef __attribute__((ext_vector_type(2))) float v2f;
typedef __attribute__((ext_vector_type(8))) float v8f;

// ---------------------------------------------------------------------------
// Kernel 1: ball query via V_WMMA_F32_16X16X4_F32.
// One wave handles 16 queries; loops over parent tiles of 16.
// d2[m][n] = |c_m|^2 + |p_n|^2 - 2 * dot(c_m, p_n); dot tile via WMMA (K=3,
// padded to 4 with zeros).
// ---------------------------------------------------------------------------
__global__ __launch_bounds__(128) void ballquery_wmma_kernel(
    const float* __restrict__ xyz,        // (BS, NP, 3)
    const float* __restrict__ child_xyz,  // (BS, MQ, 3)
    int* __restrict__ idxbuf,             // (BS, MQ, KK) workspace
    float* __restrict__ filled_out)       // (BS, MQ, KK) floats (0/1)
{
    __shared__ int s_list[4][16][KK];     // per-wave per-query index lists

    const int lane  = threadIdx.x & 31;
    const int wave  = threadIdx.x >> 5;
    const int gwave = blockIdx.x * 4 + wave;
    const int b     = gwave / (MQ / 16);
    const int qbase = (gwave % (MQ / 16)) * 16;
    const int half  = lane >> 4;          // 0: K0/K1 slice, 1: K2/K3 slice
    const int l     = lane & 15;

    // Child (query) coords for this lane's column/row
    const float* cp = child_xyz + ((size_t)b * MQ + qbase + l) * 3;
    const float cx = cp[0], cy = cp[1], cz = cp[2];
    const float cn = cx * cx + cy * cy + cz * cz;

    // A matrix 16x4 (M=query, K=coord, K3=0):
    // lanes 0-15: v0=K0(x), v1=K1(y); lanes 16-31: v0=K2(z), v1=K3(0)
    v2f A;
    A.x = half ? cz : cx;
    A.y = half ? 0.0f : cy;

    // Hoisted: |c|^2 for the query owning accumulator VGPR v in this lane
    // (query M = v + 8*half lives in lane M of the wave).
    float cnq[8];
    #pragma unroll
    for (int v = 0; v < 8; ++v)
        cnq[v] = __shfl(cn, v + (half << 3), 32);

    int count = 0;  // meaningful for lanes 0..15 (query qbase+l)

    for (int pt = 0; pt < NP / 16; ++pt) {
        const int pbase = pt * 16;
        const float* pp = xyz + ((size_t)b * NP + pbase + l) * 3;
        const float px = pp[0], py = pp[1], pz = pp[2];
        const float pn = px * px + py * py + pz * pz;

        // B matrix 4x16 (K=coord, N=parent), mirrored layout
        v2f Bm;
        Bm.x = half ? pz : px;
        Bm.y = half ? 0.0f : py;

        v8f acc = {};
        acc = __builtin_amdgcn_wmma_f32_16x16x4_f32(
            /*neg_a=*/false, A, /*neg_b=*/false, Bm,
            /*c_mod=*/(short)0, acc, /*reuse_a=*/false, /*reuse_b=*/false);

        // Each acc[v] = dot(c_{v+8*half}, p_{lane&15}).
        // Ballots are wave-uniform; keep them in registers (no LDS stash).
        unsigned bb[8];
        #pragma unroll
        for (int v = 0; v < 8; ++v) {
            const float d2 = cnq[v] + pn - 2.0f * acc[v];
            bb[v] = (unsigned)__ballot(d2 < R2);
        }

        // Lane l (<16) selects the ballot holding its query's bits via a
        // cndmask chain, then extracts its 16-bit within-ball mask.
        unsigned bw = bb[0];
        #pragma unroll
        for (int v = 1; v < 8; ++v)
            bw = ((l & 7) == v) ? bb[v] : bw;
        unsigned mymask = (l < 8) ? (bw & 0xFFFFu) : (bw >> 16);

        if (lane < 16) {
            while (mymask && count < KK) {
                const int t = __builtin_ctz(mymask);
                s_list[wave][l][count++] = pbase + t;
                mymask &= mymask - 1;
            }
        }

        // Uniform early exit when all 16 queries of this wave are full.
        const bool done = (lane >= 16) || (count >= KK);
        if ((unsigned)__ballot(done) == 0xFFFFFFFFu) break;
    }

    if (lane < 16) {
        const int q = qbase + l;
        const int firstidx = (count > 0) ? s_list[wave][l][0] : 0;
        const size_t obase = ((size_t)b * MQ + q) * KK;
        int*   ib = idxbuf + obase;
        float* fo = filled_out + obase;
        #pragma unroll 4
        for (int s = 0; s < KK; ++s) {
            const bool f = (s < count);
            ib[s] = f ? s_list[wave][l][s] : firstidx;
            fo[s] = f ? 1.0f : 0.0f;
        }
    }
}

// ---------------------------------------------------------------------------
// Kernel 2: grouping gather. One block per (b, m); 32 lanes over slots,
// 8 thread-rows over channels. Writes coalesced along the k=32 axis.
// out layout: (BS, CH, MQ, KK), channels 0..2 = relative xyz, 3.. = feats.
// ---------------------------------------------------------------------------
__global__ __launch_bounds__(256) void group_gather_kernel(
    const float* __restrict__ xyz,        // (BS, NP, 3)
    const float* __restrict__ child_xyz,  // (BS, MQ, 3)
    const float* __restrict__ feats,      // (BS, CF, NP)
    const int* __restrict__ idxbuf,       // (BS, MQ, KK)
    float* __restrict__ out)              // (BS, CH, MQ, KK)
{
    const int bm = blockIdx.x;
    const int b  = bm / MQ;
    const int m  = bm % MQ;
    const int s    = threadIdx.x & 31;
    const int crow = threadIdx.x >> 5;    // 0..7

    __shared__ int   s_idx[KK];
    __shared__ float s_c[3];
    if (threadIdx.x < KK)
        s_idx[threadIdx.x] = idxbuf[((size_t)b * MQ + m) * KK + threadIdx.x];
    if (threadIdx.x < 3)
        s_c[threadIdx.x] = child_xyz[((size_t)b * MQ + m) * 3 + threadIdx.x];
    __syncthreads();

    const int j = s_idx[s];
    const size_t outbase = (((size_t)b * CH) * MQ + m) * (size_t)KK + s;

    // relative xyz channels (3)
    if (crow < 3) {
        const float v = xyz[((size_t)b * NP + j) * 3 + crow] - s_c[crow];
        out[outbase + (size_t)crow * MQ * KK] = v;
    }

    // feature channels (128), 8 rows x 16 iterations
    const float* fb = feats + (size_t)b * CF * NP;
    #pragma unroll 4
    for (int c = crow; c < CF; c += 8) {
        const float v = fb[(size_t)c * NP + j];
        out[outbase + (size_t)(3 + c) * MQ * KK] = v;
    }
}

extern "C" void kernel_launch(void* const* d_in, const int* in_sizes, int n_in,
                              void* d_out, int out_size, void* d_ws, size_t ws_size,
                              hipStream_t stream) {
    const float* xyz       = (const float*)d_in[0];  // (2,16384,3)
    const float* child_xyz = (const float*)d_in[1];  // (2,4096,3)
    const float* feats     = (const float*)d_in[2];  // (2,128,16384)

    float* out    = (float*)d_out;                             // grouped_feats
    float* filled = out + (size_t)BS * CH * MQ * KK;           // is_filled tail
    int*   idxbuf = (int*)d_ws;                                // 1 MB scratch

    // 512 waves total (16 queries each), 4 waves per block
    ballquery_wmma_kernel<<<(BS * MQ / 16) / 4, 128, 0, stream>>>(
        xyz, child_xyz, idxbuf, filled);

    group_gather_kernel<<<BS * MQ, 256, 0, stream>>>(
        xyz, child_xyz, feats, idxbuf, out);
}